// PatchMMD_11192684774033
// MI455X (gfx1250) — compile-verified
//
#include <hip/hip_runtime.h>
#include <math.h>

#define PATCH 7
#define NIMG  64
#define CH    3
#define HW    64
#define OUTD  58                 // 64 - 7 + 1
#define NP    (OUTD * OUTD)      // 3364 patch positions
#define KDIM  147                // 3 * 7 * 7
#define KPAD  148                // pad to multiple of 4 for K=4 WMMA steps
#define SIGMA 12.0f

typedef __attribute__((ext_vector_type(2))) float v2f;
typedef __attribute__((ext_vector_type(8))) float v8f;

// ---------------------------------------------------------------------------
// Kernel 1: per-image patch norms  SA[i,p] = sum over patch of pixel^2
// ---------------------------------------------------------------------------
__global__ void patch_norm_kernel(const float* __restrict__ x,
                                  const float* __restrict__ y,
                                  float* __restrict__ sax,
                                  float* __restrict__ say) {
  int idx = blockIdx.x * blockDim.x + threadIdx.x;
  if (idx >= NIMG * NP) return;
  const float* img = (blockIdx.y == 0) ? x : y;
  float*       out = (blockIdx.y == 0) ? sax : say;
  int i  = idx / NP;
  int p  = idx - i * NP;
  int ph = p / OUTD, pw = p - (p / OUTD) * OUTD;
  const float* base = img + (size_t)i * (CH * HW * HW);
  float s = 0.f;
  for (int c = 0; c < CH; ++c) {
    for (int dh = 0; dh < PATCH; ++dh) {
      const float* row = base + c * (HW * HW) + (ph + dh) * HW + pw;
#pragma unroll
      for (int dw = 0; dw < PATCH; ++dw) { float v = row[dw]; s += v * v; }
    }
  }
  out[idx] = s;
}

// ---------------------------------------------------------------------------
// Kernel 2: per patch position, G = A_p(64x147) * B_p^T(147x64) via
// V_WMMA_F32_16X16X4_F32, fused with exp/mask/reduce.
// grid = (3364, 3 combos), block = 512 threads = 16 waves = 16 tiles of 16x16.
// ---------------------------------------------------------------------------
__global__ void mmd_wmma_kernel(const float* __restrict__ x,
                                const float* __restrict__ y,
                                const float* __restrict__ sax,
                                const float* __restrict__ say,
                                double* __restrict__ acc) {
  extern __shared__ float smem[];          // 2 * NIMG * KPAD floats (75.8 KB)
  float* Al = smem;
  float* Bl = smem + NIMG * KPAD;

  const int p     = blockIdx.x;
  const int combo = blockIdx.y;            // 0: xx, 1: xy, 2: yy
  const int ph = p / OUTD, pw = p - (p / OUTD) * OUTD;

  const float* A  = (combo == 2) ? y : x;
  const float* B  = (combo == 0) ? x : y;
  const float* SA = (combo == 2) ? say : sax;
  const float* SB = (combo == 0) ? sax : say;
  const bool excl = (combo != 1);

  const int tid = threadIdx.x;

  // Stage both 64x147 patch slabs into LDS, k = c*49 + dh*7 + dw, zero-pad k=147.
  for (int t = tid; t < NIMG * KPAD; t += blockDim.x) {
    int img = t / KPAD, k = t - img * KPAD;
    float va = 0.f, vb = 0.f;
    if (k < KDIM) {
      int c  = k / 49, r = k - c * 49;
      int dh = r / PATCH, dw = r - dh * PATCH;
      size_t off = (size_t)img * (CH * HW * HW) + (size_t)c * (HW * HW)
                 + (size_t)(ph + dh) * HW + (pw + dw);
      va = A[off];
      vb = B[off];
    }
    Al[t] = va;
    Bl[t] = vb;
  }
  __syncthreads();

  const int wave = tid >> 5;
  const int lane = tid & 31;
  const int mi   = wave >> 2;              // tile row   (i block)
  const int nj   = wave & 3;               // tile col   (j block)
  const int half = lane >> 4;              // K-half selector within fragment
  const int ln   = lane & 15;              // M (for A frag) / N (for B frag)

  const float* arow = Al + (size_t)(mi * 16 + ln) * KPAD;  // A row, this M
  const float* brow = Bl + (size_t)(nj * 16 + ln) * KPAD;  // B image, this N

  v8f c = {};
#pragma unroll 4
  for (int k0 = 0; k0 < KPAD; k0 += 4) {
    int kb = k0 + half * 2;
    v2f a = *(const v2f*)(arow + kb);      // A: VGPR0=K(kb), VGPR1=K(kb+1)
    v2f b = *(const v2f*)(brow + kb);      // B: same striping over N lanes
    c = __builtin_amdgcn_wmma_f32_16x16x4_f32(
            /*neg_a=*/false, a, /*neg_b=*/false, b,
            /*c_mod=*/(short)0, c, /*reuse_a=*/false, /*reuse_b=*/false);
  }

  // Epilogue: D = SA[i] + SB[j] - 2G, kv = exp(-D/(2 sigma^2)), mask diagonal.
  const float inv2s2 = 1.0f / (2.0f * SIGMA * SIGMA);
  const int   jcol   = nj * 16 + ln;       // N = ln for both lane halves
  const float sbj    = SB[(size_t)jcol * NP + p];
  float accf = 0.f;
#pragma unroll
  for (int r = 0; r < 8; ++r) {
    int   irow = mi * 16 + half * 8 + r;   // C/D layout: M = r (+8 on hi half)
    float sai  = SA[(size_t)irow * NP + p];
    float d    = sai + sbj - 2.0f * c[r];
    float kv   = __expf(-d * inv2s2);
    if (excl && irow == jcol) kv = 0.f;
    accf += kv;
  }

  // Block reduction (reuse LDS), then one double atomic per workgroup.
  __syncthreads();
  smem[tid] = accf;
  __syncthreads();
  for (int s = blockDim.x >> 1; s > 0; s >>= 1) {
    if (tid < (unsigned)s) smem[tid] += smem[tid + s];
    __syncthreads();
  }
  if (tid == 0) atomicAdd(&acc[combo], (double)smem[0]);
}

// ---------------------------------------------------------------------------
// Init / finalize
// ---------------------------------------------------------------------------
__global__ void init_acc_kernel(double* acc) {
  if (threadIdx.x < 3) acc[threadIdx.x] = 0.0;
}

__global__ void finalize_kernel(const double* __restrict__ acc,
                                float* __restrict__ out) {
  if (threadIdx.x == 0) {
    double offd = (double)NP * NIMG * (NIMG - 1);
    double alld = (double)NP * NIMG * NIMG;
    out[0] = (float)(acc[0] / offd - 2.0 * (acc[1] / alld) + acc[2] / offd);
  }
}

// ---------------------------------------------------------------------------
extern "C" void kernel_launch(void* const* d_in, const int* in_sizes, int n_in,
                              void* d_out, int out_size, void* d_ws,
                              size_t ws_size, hipStream_t stream) {
  (void)in_sizes; (void)n_in; (void)out_size; (void)ws_size;
  const float* x = (const float*)d_in[0];
  const float* y = (const float*)d_in[1];

  char*   ws  = (char*)d_ws;
  double* acc = (double*)ws;                       // 3 doubles
  float*  sax = (float*)(ws + 64);                 // NIMG*NP floats
  float*  say = sax + (size_t)NIMG * NP;           // NIMG*NP floats

  init_acc_kernel<<<1, 32, 0, stream>>>(acc);

  {
    int  n = NIMG * NP;
    dim3 g((n + 255) / 256, 2);
    patch_norm_kernel<<<g, 256, 0, stream>>>(x, y, sax, say);
  }
  {
    dim3   g(NP, 3);
    size_t lds = (size_t)2 * NIMG * KPAD * sizeof(float);  // 75,776 B
    mmd_wmma_kernel<<<g, 512, lds, stream>>>(x, y, sax, say, acc);
  }
  finalize_kernel<<<1, 32, 0, stream>>>(acc, (float*)d_out);
}